// AttentiveFPPoolingPyG_72164040507911
// MI455X (gfx1250) — compile-verified
//
#include <hip/hip_runtime.h>
#include <math.h>

typedef __attribute__((ext_vector_type(2))) float v2f;
typedef __attribute__((ext_vector_type(8))) float v8f;

#define N_NODES 200000
#define B_GRAPH 1024
#define H_DIM   256
#define OUT_DIM 128
#define NEG_SLOPE 0.01f

__device__ __forceinline__ float sigmoidf_(float x) { return 1.0f / (1.0f + __expf(-x)); }

// ---------------------------------------------------------------------------
// w_att[k] = sum_j W[k,j] * att_src[j]   (one block, 256 threads)
__global__ void k_watt(const float* __restrict__ W, const float* __restrict__ att_src,
                       float* __restrict__ w_att) {
    int k = threadIdx.x;
    float s = 0.0f;
    for (int j = 0; j < H_DIM; ++j) s += W[k * H_DIM + j] * att_src[j];
    w_att[k] = s;
}

// ---------------------------------------------------------------------------
// CSR segment offsets from sorted batch[]
__global__ void k_seg(const int* __restrict__ batch, int* __restrict__ seg, int n, int b) {
    int i = blockIdx.x * blockDim.x + threadIdx.x;
    if (i >= n) return;
    int cur = batch[i];
    if (i == 0) {
        for (int q = 0; q <= cur; ++q) seg[q] = 0;
    } else {
        int prev = batch[i - 1];
        for (int q = prev + 1; q <= cur; ++q) seg[q] = i;
    }
    if (i == n - 1) {
        for (int q = cur + 1; q <= b; ++q) seg[q] = n;
    }
}

// ---------------------------------------------------------------------------
// Fused: out0[b,:] = segment_sum(x) ; a_src[i] = x[i,:] . w_att
// One block (8 waves) per graph. Wave w handles node i0+w; lane covers 8 cols.
__global__ void k_pool_asrc(const float* __restrict__ x, const int* __restrict__ seg,
                            const float* __restrict__ w_att,
                            float* __restrict__ out0, float* __restrict__ a_src) {
    int b = blockIdx.x;
    int s = seg[b], e = seg[b + 1];
    int wave = threadIdx.x >> 5, lane = threadIdx.x & 31;
    float wv[8], acc[8];
    #pragma unroll
    for (int q = 0; q < 8; ++q) { wv[q] = w_att[lane + 32 * q]; acc[q] = 0.0f; }
    for (int i0 = s; i0 < e; i0 += 8) {
        int i = i0 + wave;
        if (i < e) {
            float dot = 0.0f;
            #pragma unroll
            for (int q = 0; q < 8; ++q) {
                float v = x[(size_t)i * H_DIM + lane + 32 * q];
                acc[q] += v;
                dot += v * wv[q];
            }
            for (int off = 16; off > 0; off >>= 1) dot += __shfl_down(dot, off, 32);
            if (lane == 0) a_src[i] = dot;
        }
    }
    __shared__ float lds[8][H_DIM];
    #pragma unroll
    for (int q = 0; q < 8; ++q) lds[wave][lane + 32 * q] = acc[q];
    __syncthreads();
    float sum = 0.0f;
    #pragma unroll
    for (int w = 0; w < 8; ++w) sum += lds[w][threadIdx.x];
    out0[(size_t)b * H_DIM + threadIdx.x] = sum;
}

// ---------------------------------------------------------------------------
// a_dst[row] = od[row,:] . att_dst     (one wave per row)
__global__ void k_rowdot(const float* __restrict__ od, const float* __restrict__ att,
                         float* __restrict__ a_dst, int rows) {
    int row = blockIdx.x * 8 + (threadIdx.x >> 5);
    int lane = threadIdx.x & 31;
    if (row >= rows) return;
    float d = 0.0f;
    #pragma unroll
    for (int q = 0; q < 8; ++q)
        d += od[(size_t)row * H_DIM + lane + 32 * q] * att[lane + 32 * q];
    for (int off = 16; off > 0; off >>= 1) d += __shfl_down(d, off, 32);
    if (lane == 0) a_dst[row] = d;
}

// ---------------------------------------------------------------------------
// Per-graph softmax statistics: emax[b], denom[b] over e = lrelu(a_src+a_dst)
__global__ void k_softmax_stats(const float* __restrict__ a_src, const float* __restrict__ a_dst,
                                const int* __restrict__ seg,
                                float* __restrict__ emax, float* __restrict__ denom) {
    int b = blockIdx.x;
    int s = seg[b], e = seg[b + 1];
    float ad = a_dst[b];
    __shared__ float red[256];
    float m = -INFINITY;
    for (int i = s + (int)threadIdx.x; i < e; i += 256) {
        float v = a_src[i] + ad;
        v = v > 0.0f ? v : NEG_SLOPE * v;
        m = fmaxf(m, v);
    }
    red[threadIdx.x] = m;
    __syncthreads();
    for (int off = 128; off > 0; off >>= 1) {
        if (threadIdx.x < (unsigned)off) red[threadIdx.x] = fmaxf(red[threadIdx.x], red[threadIdx.x + off]);
        __syncthreads();
    }
    m = red[0];
    __syncthreads();
    float sum = 0.0f;
    for (int i = s + (int)threadIdx.x; i < e; i += 256) {
        float v = a_src[i] + ad;
        v = v > 0.0f ? v : NEG_SLOPE * v;
        sum += expf(v - m);
    }
    red[threadIdx.x] = sum;
    __syncthreads();
    for (int off = 128; off > 0; off >>= 1) {
        if (threadIdx.x < (unsigned)off) red[threadIdx.x] += red[threadIdx.x + off];
        __syncthreads();
    }
    if (threadIdx.x == 0) { emax[b] = m; denom[b] = red[0]; }
}

// ---------------------------------------------------------------------------
// pooled[b,:] = sum_i alpha_i * x[i,:]
__global__ void k_wpool(const float* __restrict__ x, const float* __restrict__ a_src,
                        const float* __restrict__ a_dst, const float* __restrict__ emax,
                        const float* __restrict__ denom, const int* __restrict__ seg,
                        float* __restrict__ pooled) {
    int b = blockIdx.x;
    int s = seg[b], e = seg[b + 1];
    float ad = a_dst[b], m = emax[b], dn = denom[b];
    float inv = dn > 0.0f ? 1.0f / dn : 0.0f;
    int wave = threadIdx.x >> 5, lane = threadIdx.x & 31;
    float acc[8];
    #pragma unroll
    for (int q = 0; q < 8; ++q) acc[q] = 0.0f;
    for (int i0 = s; i0 < e; i0 += 8) {
        int i = i0 + wave;
        if (i < e) {
            float v = a_src[i] + ad;
            v = v > 0.0f ? v : NEG_SLOPE * v;
            float alpha = expf(v - m) * inv;
            #pragma unroll
            for (int q = 0; q < 8; ++q)
                acc[q] += alpha * x[(size_t)i * H_DIM + lane + 32 * q];
        }
    }
    __shared__ float lds[8][H_DIM];
    #pragma unroll
    for (int q = 0; q < 8; ++q) lds[wave][lane + 32 * q] = acc[q];
    __syncthreads();
    float sum = 0.0f;
    #pragma unroll
    for (int w = 0; w < 8; ++w) sum += lds[w][threadIdx.x];
    pooled[(size_t)b * H_DIM + threadIdx.x] = sum;
}

// ---------------------------------------------------------------------------
// WMMA fp32 GEMM: C[M,Ncol] = A[M,K] * B (+bias, +act). One 16x16 tile / wave.
// TRANSB==0: B is [K,Ncol] row-major. TRANSB==1: B is [Ncol,K] (use B^T).
// ACT: 0 = identity, 1 = ELU.  HAS_BIAS: add bias[col].
// Compile-time flags -> straight-line inner loop around v_wmma.
template <int TRANSB, int ACT, int HAS_BIAS>
__global__ void k_wmma_gemm(const float* __restrict__ A, const float* __restrict__ Bm,
                            const float* __restrict__ bias, float* __restrict__ C,
                            int K, int Ncol) {
    int wave = threadIdx.x >> 5, lane = threadIdx.x & 31;
    int gw = blockIdx.x * 8 + wave;
    int tilesN = Ncol >> 4;
    int tm = gw / tilesN, tn = gw % tilesN;
    int rowBase = tm << 4, colBase = tn << 4;
    int l = lane & 15;
    int kh = (lane >> 4) << 1;   // lanes 0-15 -> K pair 0,1 ; lanes 16-31 -> K pair 2,3

    const float* arow = A + (size_t)(rowBase + l) * K + kh;
    const float* bcol;
    if (TRANSB) bcol = Bm + (size_t)(colBase + l) * K + kh;
    else        bcol = Bm + (size_t)colBase + l;

    v8f c = {0.f, 0.f, 0.f, 0.f, 0.f, 0.f, 0.f, 0.f};
    #pragma unroll 4
    for (int k = 0; k < K; k += 4) {
        v2f a, bf;
        float2 av = *(const float2*)(arow + k);
        a[0] = av.x; a[1] = av.y;
        if (TRANSB) {
            float2 bv = *(const float2*)(bcol + k);
            bf[0] = bv.x; bf[1] = bv.y;
        } else {
            bf[0] = bcol[(size_t)(k + kh) * Ncol];
            bf[1] = bcol[(size_t)(k + kh + 1) * Ncol];
        }
        c = __builtin_amdgcn_wmma_f32_16x16x4_f32(false, a, false, bf,
                                                  (short)0, c, false, false);
    }

    int row0 = rowBase + (lane >> 4) * 8;
    int col = colBase + l;
    float bv = HAS_BIAS ? bias[col] : 0.0f;
    #pragma unroll
    for (int r = 0; r < 8; ++r) {
        float v = c[r] + bv;
        if (ACT == 1) v = v > 0.0f ? v : (expf(v) - 1.0f);   // ELU
        C[(size_t)(row0 + r) * Ncol + col] = v;
    }
}

// ---------------------------------------------------------------------------
// GRU gate combine + SiLU, in-place on out[]
__global__ void k_gru(const float* __restrict__ gi, const float* __restrict__ gh,
                      float* __restrict__ out, int total) {
    int idx = blockIdx.x * 256 + threadIdx.x;
    if (idx >= total) return;
    int b = idx >> 8, j = idx & 255;
    const float* gib = gi + (size_t)b * 3 * H_DIM;
    const float* ghb = gh + (size_t)b * 3 * H_DIM;
    float r = sigmoidf_(gib[j] + ghb[j]);
    float z = sigmoidf_(gib[H_DIM + j] + ghb[H_DIM + j]);
    float n = tanhf(gib[2 * H_DIM + j] + r * ghb[2 * H_DIM + j]);
    float o = out[idx];
    float v = (1.0f - z) * n + z * o;
    out[idx] = v * sigmoidf_(v);   // SiLU
}

// ---------------------------------------------------------------------------
extern "C" void kernel_launch(void* const* d_in, const int* in_sizes, int n_in,
                              void* d_out, int out_size, void* d_ws, size_t ws_size,
                              hipStream_t stream) {
    const float* x        = (const float*)d_in[0];
    const int*   batch    = (const int*)  d_in[1];
    const float* W        = (const float*)d_in[2];
    const float* att_src  = (const float*)d_in[3];
    const float* att_dst  = (const float*)d_in[4];
    const float* bias_gat = (const float*)d_in[5];
    const float* W_ih     = (const float*)d_in[6];
    const float* W_hh     = (const float*)d_in[7];
    const float* b_ih     = (const float*)d_in[8];
    const float* b_hh     = (const float*)d_in[9];
    const float* W_lin    = (const float*)d_in[10];
    const float* b_lin    = (const float*)d_in[11];
    float* out = (float*)d_out;

    // workspace carve (256B aligned slabs)
    char* ws = (char*)d_ws;
    auto carve = [&](size_t bytes) -> char* {
        char* p = ws;
        ws += (bytes + 255) & ~(size_t)255;
        return p;
    };
    int*   seg    = (int*)  carve((B_GRAPH + 1) * sizeof(int));
    float* w_att  = (float*)carve(H_DIM * sizeof(float));
    float* a_src  = (float*)carve((size_t)N_NODES * sizeof(float));
    float* st     = (float*)carve((size_t)B_GRAPH * H_DIM * sizeof(float));  // out state
    float* od     = (float*)carve((size_t)B_GRAPH * H_DIM * sizeof(float));
    float* a_dst  = (float*)carve(B_GRAPH * sizeof(float));
    float* emax   = (float*)carve(B_GRAPH * sizeof(float));
    float* denom  = (float*)carve(B_GRAPH * sizeof(float));
    float* pooled = (float*)carve((size_t)B_GRAPH * H_DIM * sizeof(float));
    float* h      = (float*)carve((size_t)B_GRAPH * H_DIM * sizeof(float));
    float* gi     = (float*)carve((size_t)B_GRAPH * 3 * H_DIM * sizeof(float));
    float* gh     = (float*)carve((size_t)B_GRAPH * 3 * H_DIM * sizeof(float));

    k_watt<<<1, H_DIM, 0, stream>>>(W, att_src, w_att);
    k_seg<<<(N_NODES + 255) / 256, 256, 0, stream>>>(batch, seg, N_NODES, B_GRAPH);
    k_pool_asrc<<<B_GRAPH, 256, 0, stream>>>(x, seg, w_att, st, a_src);

    for (int t = 0; t < 2; ++t) {
        // od = st @ W   (tiles = 64*16 = 1024 -> 128 blocks)
        k_wmma_gemm<0, 0, 0><<<128, 256, 0, stream>>>(st, W, nullptr, od, H_DIM, H_DIM);
        k_rowdot<<<B_GRAPH / 8, 256, 0, stream>>>(od, att_dst, a_dst, B_GRAPH);
        k_softmax_stats<<<B_GRAPH, 256, 0, stream>>>(a_src, a_dst, seg, emax, denom);
        k_wpool<<<B_GRAPH, 256, 0, stream>>>(x, a_src, a_dst, emax, denom, seg, pooled);
        // h = elu(pooled @ W + bias_gat)
        k_wmma_gemm<0, 1, 1><<<128, 256, 0, stream>>>(pooled, W, bias_gat, h, H_DIM, H_DIM);
        // gi = h @ W_ih^T + b_ih   (tiles = 64*48 = 3072 -> 384 blocks)
        k_wmma_gemm<1, 0, 1><<<384, 256, 0, stream>>>(h, W_ih, b_ih, gi, H_DIM, 3 * H_DIM);
        // gh = st @ W_hh^T + b_hh
        k_wmma_gemm<1, 0, 1><<<384, 256, 0, stream>>>(st, W_hh, b_hh, gh, H_DIM, 3 * H_DIM);
        k_gru<<<(B_GRAPH * H_DIM + 255) / 256, 256, 0, stream>>>(gi, gh, st, B_GRAPH * H_DIM);
    }

    // out = st @ W_lin + b_lin   (tiles = 64*8 = 512 -> 64 blocks)
    k_wmma_gemm<0, 0, 1><<<64, 256, 0, stream>>>(st, W_lin, b_lin, out, H_DIM, OUT_DIM);
}